// HausdorffDTLoss_40613210751065
// MI455X (gfx1250) — compile-verified
//
#include <hip/hip_runtime.h>

#define B_    8
#define C_    4
#define H_    256
#define W_    256
#define NPIX  65536          // H_*W_
#define NMASK 32             // B_*C_
#define NTOT  2097152        // B_*C_*H_*W_
#define BIGF  512.0f         // H+W, matches reference BIG
#define GP    257            // padded LDS pitch (bank-conflict-free both axes)

typedef __attribute__((ext_vector_type(2))) float v2f;
typedef __attribute__((ext_vector_type(8))) float v8f;

// ---------------------------------------------------------------------------
// Phase A: one block per (mask, edt-variant).  blockIdx 0..127:
//   maskid = blk>>1 (0..31 pred masks from x>0.5, 32..63 tgt masks from y==c)
//   blk&1==0 -> zero-set = ~mask ; blk&1==1 -> zero-set = mask
// Computes sqrt(edt_sq(zset)) for the whole 256x256 image and writes it to
// sqrtbuf + blk*NPIX.  The entire g image lives in LDS (CDNA5: 320KB/WGP).
// ---------------------------------------------------------------------------
__global__ __launch_bounds__(1024)
void hd_edt_half(const float* __restrict__ x, const int* __restrict__ y,
                 float* __restrict__ sqrtbuf, float* __restrict__ anyflags)
{
    __shared__ float        g[H_ * GP];       // 263,168 B
    __shared__ unsigned int mbits[NPIX / 32]; //   8,192 B (bit-packed mask)
    __shared__ int          s_any;

    const int t      = threadIdx.x;
    const int bm     = blockIdx.x;            // 0..127
    const int maskid = bm >> 1;               // 0..63
    const unsigned int inv = ((bm & 1) == 0) ? 0xFFFFFFFFu : 0u; // even: ~mask
    const int kind = maskid >> 5;             // 0: pred, 1: tgt
    const int m    = maskid & 31;
    const int bb   = m >> 2;
    const int cc   = m & 3;

    if (t == 0) s_any = 0;
    __syncthreads();

    // Bit-pack the (non-inverted) mask via wave32 ballot; coalesced loads.
    const float* xm = x + (size_t)m * NPIX;
    const int*   ym = y + (size_t)bb * NPIX;
    unsigned int anyloc = 0;
    for (int w = 0; w < NPIX / 1024; ++w) {
        const int p = (w << 10) + t;
        const bool bit = (kind == 0) ? (xm[p] > 0.5f) : (ym[p] == cc);
        const unsigned int bal = __builtin_amdgcn_ballot_w32(bit);
        anyloc |= bal;
        if ((t & 31) == 0) mbits[p >> 5] = bal;
    }
    if (anyloc) atomicOr(&s_any, 1);
    __syncthreads();

    if (t == 0) anyflags[maskid] = s_any ? 1.0f : 0.0f; // both half-blocks write same value

    // ---- horizontal 1D distance (one thread per row), g = min(df,db)^2 ----
    if (t < H_) {
        const int r = t;
        int last = -1;
        for (int wi = 0; wi < 8; ++wi) {
            const unsigned int wv = mbits[(r << 3) + wi] ^ inv;
            for (int bj = 0; bj < 32; ++bj) {
                const int j = (wi << 5) + bj;
                if ((wv >> bj) & 1u) last = j;
                g[r * GP + j] = (last >= 0) ? (float)(j - last) : BIGF;
            }
        }
        int nxt = -1;
        for (int wi = 7; wi >= 0; --wi) {
            const unsigned int wv = mbits[(r << 3) + wi] ^ inv;
            for (int bj = 31; bj >= 0; --bj) {
                const int j = (wi << 5) + bj;
                if ((wv >> bj) & 1u) nxt = j;
                const float db = (nxt >= 0) ? (float)(nxt - j) : BIGF;
                const float d  = fminf(g[r * GP + j], db);
                g[r * GP + j]  = d * d;
            }
        }
    }
    __syncthreads();

    // ---- vertical min-plus: D2[i,j] = min_ip g[ip,j] + (i-ip)^2 ----
    // thread -> fixed column j, 64 output rows, processed 8 at a time so each
    // ds_load of g feeds 8 fma+add+min chains.
    float* sout = sqrtbuf + (size_t)bm * NPIX;
    const int j     = t & (W_ - 1);
    const int ibase = (t >> 8) << 6;          // 0,64,128,192
    for (int c8 = 0; c8 < 8; ++c8) {
        const int i0 = ibase + (c8 << 3);
        float acc[8];
#pragma unroll
        for (int u = 0; u < 8; ++u) acc[u] = 3.0e38f;
        for (int ip = 0; ip < H_; ++ip) {
            const float gv  = g[ip * GP + j];
            const float fb  = (float)(i0 - ip);
            const float fb2 = fb * fb;
#pragma unroll
            for (int u = 0; u < 8; ++u) {
                const float off = fmaf(fb, (float)(2 * u), fb2 + (float)(u * u));
                acc[u] = fminf(acc[u], gv + off);
            }
        }
#pragma unroll
        for (int u = 0; u < 8; ++u)
            sout[(i0 + u) * W_ + j] = sqrtf(acc[u]);  // coalesced (lanes = adjacent j)
    }
}

// ---------------------------------------------------------------------------
// Phase B: fused field-combine + weighted-error + reduction.
// Per-lane values are folded through V_WMMA_F32_16X16X4_F32 with B = ones:
// D += A x ones accumulates exact-f32 row sums (replicated over 16 cols).
// 512 blocks x 256 threads x 16 elems = NTOT exactly; no divergence around
// the WMMA (EXEC must be all ones).
// ---------------------------------------------------------------------------
__global__ __launch_bounds__(256)
void hd_wsum(const float* __restrict__ x, const int* __restrict__ y,
             const float* __restrict__ sqrtbuf, const float* __restrict__ anyflags,
             float* __restrict__ partials)
{
    __shared__ float sdata[256];
    const int t    = threadIdx.x;
    const int blk  = blockIdx.x;
    const int base = blk * 4096;

    v8f cacc = {0.f, 0.f, 0.f, 0.f, 0.f, 0.f, 0.f, 0.f};
    const v2f bones = {1.0f, 1.0f};

#pragma unroll
    for (int k = 0; k < 8; ++k) {
        float v[2];
#pragma unroll
        for (int h = 0; h < 2; ++h) {
            const int e  = base + ((2 * k + h) << 8) + t;   // lane-consecutive
            const int mm = e >> 16;                          // b*C+c
            const int hw = e & (NPIX - 1);
            const float xv = x[e];
            const float oh = (y[((mm >> 2) << 16) + hw] == (mm & 3)) ? 1.0f : 0.0f;
            const size_t pb = ((size_t)(mm << 1)) << 16;
            const size_t tb = ((size_t)((NMASK + mm) << 1)) << 16;
            const float pd = (sqrtbuf[pb + hw] + sqrtbuf[pb + NPIX + hw]) * anyflags[mm];
            const float td = (sqrtbuf[tb + hw] + sqrtbuf[tb + NPIX + hw]) * anyflags[NMASK + mm];
            const float diff = xv - oh;
            v[h] = diff * diff * (pd * pd + td * td);        // ALPHA = 2
        }
        v2f a = {v[0], v[1]};
        // D = A x ones + C  (exact f32 adds, RNE) -> row-sum accumulator
        cacc = __builtin_amdgcn_wmma_f32_16x16x4_f32(false, a, false, bones,
                                                     (short)0, cacc, false, false);
    }

    float s = 0.0f;
#pragma unroll
    for (int u = 0; u < 8; ++u) s += cacc[u];
    sdata[t] = s;
    __syncthreads();
    for (int st = 128; st > 0; st >>= 1) {
        if (t < st) sdata[t] += sdata[t + st];
        __syncthreads();
    }
    // /16: ones-B replicates every row sum across 16 columns of D
    if (t == 0) partials[blk] = sdata[0] * 0.0625f;
}

__global__ __launch_bounds__(256)
void hd_finalize(const float* __restrict__ partials, float* __restrict__ out)
{
    __shared__ float sdata[256];
    const int t = threadIdx.x;
    sdata[t] = partials[t] + partials[t + 256];
    __syncthreads();
    for (int st = 128; st > 0; st >>= 1) {
        if (t < st) sdata[t] += sdata[t + st];
        __syncthreads();
    }
    if (t == 0) out[0] = sdata[0] * (1.0f / (float)NTOT);   // mean
}

// ---------------------------------------------------------------------------
extern "C" void kernel_launch(void* const* d_in, const int* in_sizes, int n_in,
                              void* d_out, int out_size, void* d_ws, size_t ws_size,
                              hipStream_t stream)
{
    (void)in_sizes; (void)n_in; (void)out_size; (void)ws_size;
    const float* x = (const float*)d_in[0];   // [8,4,256,256] f32
    const int*   y = (const int*)d_in[1];     // [8,256,256] i32

    // workspace layout: 128 sqrt-fields (33.55 MB) | 64 any-flags | 512 partials
    float* sqrtbuf  = (float*)d_ws;
    float* anyflags = sqrtbuf + (size_t)128 * NPIX;
    float* partials = anyflags + 64;

    hd_edt_half<<<128, 1024, 0, stream>>>(x, y, sqrtbuf, anyflags);
    hd_wsum   <<<512,  256, 0, stream>>>(x, y, sqrtbuf, anyflags, partials);
    hd_finalize<<<1,   256, 0, stream>>>(partials, (float*)d_out);
}